// TemporalGCN_11879879544625
// MI455X (gfx1250) — compile-verified
//
#include <hip/hip_runtime.h>
#include <hip/hip_bf16.h>

// ---------------------------------------------------------------------------
// TemporalGCN forward for MI455X (gfx1250, wave32, WMMA).
// All GEMMs: V_WMMA_F32_16X16X4_F32, one wave owns a 16(M)x64(N) tile.
// Weights are staged once per 256-thread block into LDS as zero-padded
// K-pair float2 (ds_load_b64 in the hot loop, shared by 8 waves).
// A operands: contiguous rows -> direct global_load_b64 (float2); irregular
// K (semb 55, GCN1 concat 69) -> per-wave zero-padded LDS tile, keeping the
// MMA loop branch-free (WMMA requires EXEC all-ones; only addresses diverge).
// Working set (~200MB) is L2-resident (192MB); GCN aggregation uses
// global_atomic_add_f32. LSTM = 16 sequential (WMMA gates + elementwise).
// ---------------------------------------------------------------------------

typedef __attribute__((ext_vector_type(2))) float v2f;
typedef __attribute__((ext_vector_type(8))) float v8f;

namespace {
constexpr int cB = 2, cT = 16, cN = 10000, cF = 5, cS = 55, cE = 80000;
constexpr int cH = 64, cFUT = 24;
constexpr int cBT = cB * cT;   // 32
constexpr int cBN = cB * cN;   // 20000
constexpr int cH4 = 4 * cH;    // 256
constexpr int WPB = 8;         // waves (m-tiles) per 256-thread block
}

// ISA f32 layouts (CDNA5 ISA §7.12.2):
//   A 16x4 : lane(0..15)->M=lane&15, VGPR0/1 = K=0/1 ; lanes 16..31 = K=2/3
//   B 4x16 : VGPR0/1 hold rows K=(2*kh)/(2*kh+1), col = lane&15
//   C/D    : VGPR r : M = r + 8*(lane>>4), N = lane&15
__device__ __forceinline__ v8f wmma4(v2f a, v2f b, v8f c) {
  return __builtin_amdgcn_wmma_f32_16x16x4_f32(
      /*neg_a=*/false, a, /*neg_b=*/false, b,
      /*c_mod=*/(short)0, c, /*reuse_a=*/false, /*reuse_b=*/false);
}

// Stage B[K x 64] into LDS as K-pair float2: sB[pa*64 + c] = {B(2pa,c),B(2pa+1,c)}
template <int KP, class BSrc>
__device__ __forceinline__ void fill_B(v2f* sB, BSrc src) {
  for (int idx = threadIdx.x; idx < (KP / 2) * 64; idx += blockDim.x) {
    int pa = idx >> 6, c = idx & 63;
    v2f v;
    v.x = src(2 * pa, c);
    v.y = src(2 * pa + 1, c);
    sB[idx] = v;
  }
}

// Branch-free MMA loop: uniform trip count, divergent k-pair pa = 2*s + kh.
template <int KP, class APair>
__device__ __forceinline__ void wave_mma(APair ap, const v2f* sB, v8f* acc,
                                         int row, int kh) {
  for (int s = 0; s < KP / 4; ++s) {
    const int pa = 2 * s + kh;
    v2f a = ap(pa);
#pragma unroll
    for (int j = 0; j < 4; ++j) {
      v2f b = sB[pa * 64 + j * 16 + row];
      acc[j] = wmma4(a, b, acc[j]);
    }
  }
}

// ---------------------------------------------------------------------------
// Utility kernels
// ---------------------------------------------------------------------------
__global__ void k_fill(float* __restrict__ p, float v, long long n) {
  long long i = (long long)blockIdx.x * blockDim.x + threadIdx.x;
  if (i < n) p[i] = v;
}

__global__ void k_deg(const int* __restrict__ graph, float* __restrict__ deg) {
  int e = blockIdx.x * blockDim.x + threadIdx.x;
  if (e < cE) atomicAdd(&deg[graph[cE + e]], 1.0f);  // dst row of graph
}

__global__ void k_dinv(const float* __restrict__ deg, float* __restrict__ dinv) {
  int n = blockIdx.x * blockDim.x + threadIdx.x;
  if (n < cN) dinv[n] = rsqrtf(deg[n]);  // deg >= 1 (self loops)
}

__global__ void k_bias_relu(float* __restrict__ p, const float* __restrict__ bias,
                            long long n) {
  long long i = (long long)blockIdx.x * blockDim.x + threadIdx.x;
  if (i < n) {
    float v = p[i] + bias[i % cH];
    p[i] = v > 0.0f ? v : 0.0f;
  }
}

// ---------------------------------------------------------------------------
// semb = static @ static_W + static_b   [N,55] x [55,64] -> [N,64]   KP=56
// ---------------------------------------------------------------------------
__global__ void k_semb(const float* __restrict__ stat, const float* __restrict__ W,
                       const float* __restrict__ bias, float* __restrict__ semb) {
  constexpr int KP = 56;
  __shared__ v2f sB[(KP / 2) * 64];
  __shared__ v2f sA[WPB][(KP / 2) * 16];
  const int wv = threadIdx.x >> 5, lane = threadIdx.x & 31;
  const int row = lane & 15, kh = lane >> 4;
  const int m0 = (blockIdx.x * WPB + wv) * 16;

  fill_B<KP>(sB, [&](int k, int c) -> float {
    return (k < cS) ? W[(size_t)k * cH + c] : 0.0f;
  });
  for (int idx = lane; idx < (KP / 2) * 16; idx += 32) {
    int pa = idx >> 4, r = idx & 15, m = m0 + r;
    v2f v;
    v.x = (2 * pa < cS && m < cN) ? stat[(size_t)m * cS + 2 * pa] : 0.0f;
    v.y = (2 * pa + 1 < cS && m < cN) ? stat[(size_t)m * cS + 2 * pa + 1] : 0.0f;
    sA[wv][idx] = v;
  }
  __syncthreads();

  v8f acc[4] = {};
  wave_mma<KP>([&](int pa) { return sA[wv][pa * 16 + row]; }, sB, acc, row, kh);
#pragma unroll
  for (int j = 0; j < 4; ++j)
#pragma unroll
    for (int r = 0; r < 8; ++r) {
      int m = m0 + r + 8 * kh, c = j * 16 + row;
      if (m < cN) semb[(size_t)m * cH + c] = acc[j][r] + bias[c];
    }
}

// ---------------------------------------------------------------------------
// GCN1: xw = [X | semb] @ gcn1_W   [BT*N,69] x [69,64]   KP=72
// ---------------------------------------------------------------------------
__global__ void k_gcn1_gemm(const float* __restrict__ X, const float* __restrict__ semb,
                            const float* __restrict__ W, float* __restrict__ out) {
  constexpr int K = cF + cH;  // 69
  constexpr int KP = 72;
  __shared__ v2f sB[(KP / 2) * 64];
  __shared__ v2f sA[WPB][(KP / 2) * 16];
  const int wv = threadIdx.x >> 5, lane = threadIdx.x & 31;
  const int row = lane & 15, kh = lane >> 4;
  const int m0 = (blockIdx.x * WPB + wv) * 16;  // tiles exact: 320000/16

  fill_B<KP>(sB, [&](int k, int c) -> float {
    return (k < K) ? W[(size_t)k * cH + c] : 0.0f;
  });
  auto aval = [&](int m, int k) -> float {
    if (k >= K) return 0.0f;
    if (k < cF) return X[(size_t)m * cF + k];
    return semb[(size_t)(m % cN) * cH + (k - cF)];
  };
  for (int idx = lane; idx < (KP / 2) * 16; idx += 32) {
    int pa = idx >> 4, r = idx & 15, m = m0 + r;
    v2f v;
    v.x = aval(m, 2 * pa);
    v.y = aval(m, 2 * pa + 1);
    sA[wv][idx] = v;
  }
  __syncthreads();

  v8f acc[4] = {};
  wave_mma<KP>([&](int pa) { return sA[wv][pa * 16 + row]; }, sB, acc, row, kh);
#pragma unroll
  for (int j = 0; j < 4; ++j)
#pragma unroll
    for (int r = 0; r < 8; ++r) {
      int m = m0 + r + 8 * kh, c = j * 16 + row;
      out[(size_t)m * cH + c] = acc[j][r];
    }
}

// ---------------------------------------------------------------------------
// GCN2: out = in @ gcn2_W   [BT*N,64] x [64,64]   KP=64, A = direct b64 loads
// ---------------------------------------------------------------------------
__global__ void k_gcn2_gemm(const float* __restrict__ A, const float* __restrict__ W,
                            float* __restrict__ out) {
  constexpr int KP = 64;
  __shared__ v2f sB[(KP / 2) * 64];
  const int wv = threadIdx.x >> 5, lane = threadIdx.x & 31;
  const int row = lane & 15, kh = lane >> 4;
  const int m0 = (blockIdx.x * WPB + wv) * 16;  // tiles exact

  fill_B<KP>(sB, [&](int k, int c) -> float { return W[(size_t)k * cH + c]; });
  __syncthreads();

  const float* arow = A + (size_t)(m0 + row) * cH;
  v8f acc[4] = {};
  wave_mma<KP>([&](int pa) { return *(const v2f*)(arow + 2 * pa); }, sB, acc, row, kh);
#pragma unroll
  for (int j = 0; j < 4; ++j)
#pragma unroll
    for (int r = 0; r < 8; ++r) {
      int m = m0 + r + 8 * kh, c = j * 16 + row;
      out[(size_t)m * cH + c] = acc[j][r];
    }
}

// ---------------------------------------------------------------------------
// Edge scatter: agg[bt,dst] += xw[bt,src] * dinv[src]*dinv[dst]
// ---------------------------------------------------------------------------
__global__ void k_scatter(const float* __restrict__ xw, const int* __restrict__ graph,
                          const float* __restrict__ dinv, float* __restrict__ agg) {
  long long tid = (long long)blockIdx.x * blockDim.x + threadIdx.x;
  const long long total = (long long)cBT * (cE + cN) * cH;
  if (tid >= total) return;
  int hcol = (int)(tid % cH);
  long long r = tid / cH;
  int e = (int)(r % (cE + cN));
  int bt = (int)(r / (cE + cN));
  int s, d;
  if (e < cE) { s = graph[e]; d = graph[cE + e]; }
  else        { s = d = e - cE; }
  float nrm = dinv[s] * dinv[d];
  float v = xw[((size_t)bt * cN + s) * cH + hcol] * nrm;
  atomicAdd(&agg[((size_t)bt * cN + d) * cH + hcol], v);
}

// ---------------------------------------------------------------------------
// LSTM gates (one step): gates = xt @ Wih^T + h @ Whh^T
// gridDim.y selects a 64-wide slab of the 256 gate columns.
// Wih/Whh rows are contiguous in k -> pair loads in fill are b64.
// ---------------------------------------------------------------------------
__global__ void k_lstm_gates(const float* __restrict__ xseq, const float* __restrict__ h,
                             const float* __restrict__ Wih, const float* __restrict__ Whh,
                             float* __restrict__ gates, int t) {
  constexpr int KP = 64;
  __shared__ v2f sBi[(KP / 2) * 64];
  __shared__ v2f sBh[(KP / 2) * 64];
  const int wv = threadIdx.x >> 5, lane = threadIdx.x & 31;
  const int row = lane & 15, kh = lane >> 4;
  const int m0 = (blockIdx.x * WPB + wv) * 16;
  const int n0 = blockIdx.y * 64;

  fill_B<KP>(sBi, [&](int k, int c) -> float { return Wih[(size_t)(n0 + c) * cH + k]; });
  fill_B<KP>(sBh, [&](int k, int c) -> float { return Whh[(size_t)(n0 + c) * cH + k]; });
  __syncthreads();

  int rowz = m0 + row;
  if (rowz >= cBN) rowz = cBN - 1;  // clamp OOB waves of last block
  const int b = rowz / cN, n = rowz % cN;
  const float* xrow = xseq + ((size_t)(b * cT + t) * cN + n) * cH;
  const float* hrow = h + (size_t)rowz * cH;

  v8f acc[4] = {};
  wave_mma<KP>([&](int pa) { return *(const v2f*)(xrow + 2 * pa); }, sBi, acc, row, kh);
  wave_mma<KP>([&](int pa) { return *(const v2f*)(hrow + 2 * pa); }, sBh, acc, row, kh);
#pragma unroll
  for (int j = 0; j < 4; ++j)
#pragma unroll
    for (int r = 0; r < 8; ++r) {
      int m = m0 + r + 8 * kh, c = n0 + j * 16 + row;
      if (m < cBN) gates[(size_t)m * cH4 + c] = acc[j][r];
    }
}

// ---------------------------------------------------------------------------
// LSTM elementwise update (torch gate order i,f,g,o)
// ---------------------------------------------------------------------------
__global__ void k_lstm_update(const float* __restrict__ gates,
                              const float* __restrict__ bih, const float* __restrict__ bhh,
                              float* __restrict__ h, float* __restrict__ c) {
  int tid = blockIdx.x * blockDim.x + threadIdx.x;
  if (tid >= cBN * cH) return;
  int rowz = tid / cH, j = tid % cH;
  const float* g = gates + (size_t)rowz * cH4;
  float gi = g[j]          + bih[j]          + bhh[j];
  float gf = g[cH + j]     + bih[cH + j]     + bhh[cH + j];
  float gg = g[2 * cH + j] + bih[2 * cH + j] + bhh[2 * cH + j];
  float go = g[3 * cH + j] + bih[3 * cH + j] + bhh[3 * cH + j];
  float si = 1.0f / (1.0f + __expf(-gi));
  float sf = 1.0f / (1.0f + __expf(-gf));
  float so = 1.0f / (1.0f + __expf(-go));
  float cn = sf * c[tid] + si * tanhf(gg);
  c[tid] = cn;
  h[tid] = so * tanhf(cn);
}

// ---------------------------------------------------------------------------
// Decoder layer 1: d1 = relu(hT @ dec1_W + b)   [BN,64] x [64,64]
// ---------------------------------------------------------------------------
__global__ void k_dec1(const float* __restrict__ hT, const float* __restrict__ W,
                       const float* __restrict__ bias, float* __restrict__ d1) {
  constexpr int KP = 64;
  __shared__ v2f sB[(KP / 2) * 64];
  const int wv = threadIdx.x >> 5, lane = threadIdx.x & 31;
  const int row = lane & 15, kh = lane >> 4;
  const int m0 = (blockIdx.x * WPB + wv) * 16;

  fill_B<KP>(sB, [&](int k, int c) -> float { return W[(size_t)k * cH + c]; });
  __syncthreads();

  int rowz = m0 + row;
  if (rowz >= cBN) rowz = cBN - 1;
  const float* arow = hT + (size_t)rowz * cH;
  v8f acc[4] = {};
  wave_mma<KP>([&](int pa) { return *(const v2f*)(arow + 2 * pa); }, sB, acc, row, kh);
#pragma unroll
  for (int j = 0; j < 4; ++j)
#pragma unroll
    for (int r = 0; r < 8; ++r) {
      int m = m0 + r + 8 * kh, c = j * 16 + row;
      if (m < cBN) {
        float v = acc[j][r] + bias[c];
        d1[(size_t)m * cH + c] = v > 0.0f ? v : 0.0f;
      }
    }
}

// ---------------------------------------------------------------------------
// Decoder layer 2 + transposed store: out[b,fut,n] = (d1 @ dec2_W + b)[b*N+n,fut]
// [BN,64] x [64,24]; cols >= 24 zero-padded in LDS, masked on store.
// ---------------------------------------------------------------------------
__global__ void k_dec2(const float* __restrict__ d1, const float* __restrict__ W,
                       const float* __restrict__ bias, float* __restrict__ out) {
  constexpr int KP = 64;
  __shared__ v2f sB[(KP / 2) * 64];
  const int wv = threadIdx.x >> 5, lane = threadIdx.x & 31;
  const int row = lane & 15, kh = lane >> 4;
  const int m0 = (blockIdx.x * WPB + wv) * 16;

  fill_B<KP>(sB, [&](int k, int c) -> float {
    return (c < cFUT) ? W[(size_t)k * cFUT + c] : 0.0f;
  });
  __syncthreads();

  int rowz = m0 + row;
  if (rowz >= cBN) rowz = cBN - 1;
  const float* arow = d1 + (size_t)rowz * cH;
  v8f acc[4] = {};
  wave_mma<KP>([&](int pa) { return *(const v2f*)(arow + 2 * pa); }, sB, acc, row, kh);
#pragma unroll
  for (int j = 0; j < 2; ++j)  // only cols 0..31 can be valid
#pragma unroll
    for (int r = 0; r < 8; ++r) {
      int m = m0 + r + 8 * kh, c = j * 16 + row;
      if (c < cFUT && m < cBN) {
        int b = m / cN, n = m % cN;
        out[((size_t)b * cFUT + c) * cN + n] = acc[j][r] + bias[c];
      }
    }
}

// ---------------------------------------------------------------------------
// Host-side orchestration
// ---------------------------------------------------------------------------
extern "C" void kernel_launch(void* const* d_in, const int* in_sizes, int n_in,
                              void* d_out, int out_size, void* d_ws, size_t ws_size,
                              hipStream_t stream) {
  const float* X        = (const float*)d_in[0];
  const float* stat     = (const float*)d_in[1];
  const int*   graph    = (const int*)  d_in[2];
  const float* static_W = (const float*)d_in[3];
  const float* static_b = (const float*)d_in[4];
  const float* gcn1_W   = (const float*)d_in[5];
  const float* gcn1_b   = (const float*)d_in[6];
  const float* gcn2_W   = (const float*)d_in[7];
  const float* gcn2_b   = (const float*)d_in[8];
  const float* lstm_Wih = (const float*)d_in[9];
  const float* lstm_Whh = (const float*)d_in[10];
  const float* lstm_bih = (const float*)d_in[11];
  const float* lstm_bhh = (const float*)d_in[12];
  const float* dec1_W   = (const float*)d_in[13];
  const float* dec1_b   = (const float*)d_in[14];
  const float* dec2_W   = (const float*)d_in[15];
  const float* dec2_b   = (const float*)d_in[16];
  float* out = (float*)d_out;

  // Workspace carve-up (floats), total ~50.6M floats (~202MB)
  float* ws = (float*)d_ws;
  float* semb  = ws;                            // N*H
  float* deg   = semb + (size_t)cN * cH;        // N
  float* dinv  = deg + cN;                      // N
  float* buf1  = dinv + cN;                     // BT*N*H
  float* buf2  = buf1 + (size_t)cBT * cN * cH;  // BT*N*H
  float* gates = buf2 + (size_t)cBT * cN * cH;  // BN*4H
  float* hb    = gates + (size_t)cBN * cH4;     // BN*H
  float* cb    = hb + (size_t)cBN * cH;         // BN*H
  float* d1    = cb + (size_t)cBN * cH;         // BN*H
  (void)ws_size; (void)in_sizes; (void)n_in; (void)out_size;

  const long long actN = (long long)cBT * cN * cH;  // 20,480,000
  auto blks = [](long long n, int bs) { return (unsigned)((n + bs - 1) / bs); };

  // --- degree / normalization ---
  k_fill<<<blks(cN, 256), 256, 0, stream>>>(deg, 1.0f, cN);  // self loops
  k_deg<<<blks(cE, 256), 256, 0, stream>>>(graph, deg);
  k_dinv<<<blks(cN, 256), 256, 0, stream>>>(deg, dinv);

  // --- static embedding GEMM ---
  {
    int tiles = (cN + 15) / 16;  // 625
    k_semb<<<blks(tiles, WPB), 32 * WPB, 0, stream>>>(stat, static_W, static_b, semb);
  }

  // --- GCN layer 1: GEMM -> scatter -> bias+relu ---
  {
    int tiles = (cBT * cN) / 16;  // 20000
    k_gcn1_gemm<<<blks(tiles, WPB), 32 * WPB, 0, stream>>>(X, semb, gcn1_W, buf1);
  }
  k_fill<<<blks(actN, 256), 256, 0, stream>>>(buf2, 0.0f, actN);
  {
    long long total = (long long)cBT * (cE + cN) * cH;
    k_scatter<<<blks(total, 256), 256, 0, stream>>>(buf1, graph, dinv, buf2);
  }
  k_bias_relu<<<blks(actN, 256), 256, 0, stream>>>(buf2, gcn1_b, actN);

  // --- GCN layer 2: GEMM -> scatter -> bias+relu ---
  {
    int tiles = (cBT * cN) / 16;
    k_gcn2_gemm<<<blks(tiles, WPB), 32 * WPB, 0, stream>>>(buf2, gcn2_W, buf1);
  }
  k_fill<<<blks(actN, 256), 256, 0, stream>>>(buf2, 0.0f, actN);
  {
    long long total = (long long)cBT * (cE + cN) * cH;
    k_scatter<<<blks(total, 256), 256, 0, stream>>>(buf1, graph, dinv, buf2);
  }
  k_bias_relu<<<blks(actN, 256), 256, 0, stream>>>(buf2, gcn2_b, actN);
  // buf2 now holds the LSTM input sequence, laid out [b*T+t][n][h]

  // --- LSTM over T steps ---
  k_fill<<<blks((long long)cBN * cH, 256), 256, 0, stream>>>(hb, 0.0f, (long long)cBN * cH);
  k_fill<<<blks((long long)cBN * cH, 256), 256, 0, stream>>>(cb, 0.0f, (long long)cBN * cH);
  {
    int tiles = cBN / 16;  // 1250
    dim3 ggrid(blks(tiles, WPB), 4);  // 4 x 64-wide gate column slabs
    for (int t = 0; t < cT; ++t) {
      k_lstm_gates<<<ggrid, 32 * WPB, 0, stream>>>(buf2, hb, lstm_Wih, lstm_Whh, gates, t);
      k_lstm_update<<<blks((long long)cBN * cH, 256), 256, 0, stream>>>(
          gates, lstm_bih, lstm_bhh, hb, cb);
    }
  }

  // --- Decoder ---
  {
    int tiles = cBN / 16;
    k_dec1<<<blks(tiles, WPB), 32 * WPB, 0, stream>>>(hb, dec1_W, dec1_b, d1);
    k_dec2<<<blks(tiles, WPB), 32 * WPB, 0, stream>>>(d1, dec2_W, dec2_b, out);
  }
}